// SpikingNet_61168924229667
// MI455X (gfx1250) — compile-verified
//
#include <hip/hip_runtime.h>
#include <stdint.h>

// ---------------------------------------------------------------------------
// Recurrent spiking net on MI455X (gfx1250).
// Persistent kernel: 1 workgroup per 16 batch rows, loops T=100 steps with all
// recurrent state (syn1, mem1, mem2) in WMMA accumulator VGPRs, spikes in LDS.
// Matrix math via V_WMMA_F32_16X16X4_F32 (fp32 matches reference numerics;
// compute is not the bottleneck -- the serial T-loop is).
// x tiles are DMA'd by the Tensor Data Mover one step ahead (double-buffered)
// so the DMA overlaps the GEMM phase of the previous step.
// ---------------------------------------------------------------------------

typedef float v2f  __attribute__((ext_vector_type(2)));
typedef float v8f  __attribute__((ext_vector_type(8)));
typedef unsigned int u32x4 __attribute__((ext_vector_type(4)));
typedef int   i32x4 __attribute__((ext_vector_type(4)));
typedef int   i32x8 __attribute__((ext_vector_type(8)));

#define T_STEPS 100
#define BATCH   512
#define NIN     256
#define NH      512
#define NOUT    256
#define ROWS    16            // batch rows per workgroup
#define NWAVES  8             // waves per workgroup (wave32)
#define BLOCK   (NWAVES * 32)
#define XSTRIDE 260           // x tile LDS row stride (dwords): 256 + 4 pad
#define SSTRIDE 17            // spike tile LDS row stride (dwords): 16 + 1 pad
#define ALPHA_C 0.9f
#define BETA_C  0.9f
#define THR_C   1.0f

#if defined(__has_builtin)
#  if __has_builtin(__builtin_amdgcn_tensor_load_to_lds)
#    define USE_TDM 1
#  endif
#endif
#ifndef USE_TDM
#  define USE_TDM 0
#endif

__device__ __forceinline__ v8f wmma4(v2f a, v2f b, v8f c) {
  // D = A(16x4 f32) * B(4x16 f32) + C(16x16 f32)
  return __builtin_amdgcn_wmma_f32_16x16x4_f32(
      /*neg_a=*/false, a, /*neg_b=*/false, b,
      /*c_mod=*/(short)0, c, /*reuse_a=*/false, /*reuse_b=*/false);
}

#if USE_TDM
// Issue a TDM load of a 256x16 fp32 tile (row-major, row stride 256 elements)
// into LDS at lds_off, padding 4 dwords after each 256-dword row so the LDS
// image has row stride XSTRIDE (260) dwords -> conflict-free transposed reads.
__device__ __forceinline__ void issue_x_tdm(const float* xsrc, unsigned lds_off) {
  unsigned long long ga = (unsigned long long)(size_t)xsrc;
  u32x4 g0;
  g0.x = 1u;                                   // count=1 valid descriptor
  g0.y = lds_off;                              // lds_addr (bytes)
  g0.z = (unsigned)(ga & 0xFFFFFFFFu);         // global_addr[31:0]
  g0.w = (unsigned)((ga >> 32) & 0x01FFFFFFu)  // global_addr[56:32]
         | (2u << 30);                         // type = 2 ("image")
  i32x8 g1;
  g1[0] = (int)((2u << 16)      // data_size = 4B
              | (1u << 20)      // pad_enable
              | (7u << 22)      // pad_interval: every 256 dwords
              | (3u << 25));    // pad_amount: 4 dwords
  g1[1] = (int)(256u << 16);    // tensor_dim0[15:0] = 256 (k)
  g1[2] = (int)(16u << 16);     // tensor_dim0 hi = 0 | tensor_dim1 = 16 (rows)
  g1[3] = (int)(256u << 16);    // tensor_dim1 hi = 0 | tile_dim0 = 256
  g1[4] = 16;                   // tile_dim1 = 16 | tile_dim2 = 0
  g1[5] = 256;                  // tensor_dim0_stride[31:0] = 256 elements
  g1[6] = 0;
  g1[7] = 0;
  i32x4 gz4 = {0, 0, 0, 0};
  i32x8 gz8 = {0, 0, 0, 0, 0, 0, 0, 0};
  __builtin_amdgcn_tensor_load_to_lds(g0, g1, gz4, gz4, gz8, 0);
}
#endif

// One-time transpose of W2 [NOUT, NH] -> W2T [NH, NOUT] in workspace so the
// GEMM2 B-operand reads are contiguous.
__global__ void transpose_w2_kernel(const float* __restrict__ W2,
                                    float* __restrict__ W2T) {
  int idx = blockIdx.x * blockDim.x + threadIdx.x;
  if (idx < NOUT * NH) {
    int o = idx / NH;
    int h = idx - o * NH;
    W2T[h * NOUT + o] = W2[idx];
  }
}

__global__ void __launch_bounds__(BLOCK, 1)
snn_persistent_kernel(const float* __restrict__ x,
                      const float* __restrict__ W1,
                      const float* __restrict__ Wrec,
                      const float* __restrict__ W2T,
                      float* __restrict__ spk_out,
                      float* __restrict__ mem_out)
{
  __shared__ float xT[2][ROWS * XSTRIDE];  // double-buffered x tile
  __shared__ float spkT[NH * SSTRIDE];     // spk1 tile laid out [h][b], padded

  const int tid  = threadIdx.x;
  const int lane = tid & 31;
  const int wave = tid >> 5;
  const int lm   = lane & 15;            // lane % 16
  const int lh   = lane >> 4;            // lane half (0/1)
  const int b0   = blockIdx.x * ROWS;    // this WG's batch rows

  // Register-resident recurrent state (C/D-matrix WMMA layout).
  // Wave w owns hidden units n in [w*64, w*64+64) -> 4 M-tiles of 16.
  // Wave w owns outputs o in [w*32, w*32+32)      -> 2 N-tiles of 16.
  v8f syn[4], mem1[4], mem2[2];
  #pragma unroll
  for (int i = 0; i < 4; ++i) {
    #pragma unroll
    for (int r = 0; r < 8; ++r) { syn[i][r] = 0.0f; mem1[i][r] = 0.0f; }
  }
  #pragma unroll
  for (int i = 0; i < 2; ++i) {
    #pragma unroll
    for (int r = 0; r < 8; ++r) { mem2[i][r] = 0.0f; }
  }

  // spk1 initial state = 0
  for (int i = tid; i < NH * SSTRIDE; i += BLOCK) spkT[i] = 0.0f;

#if USE_TDM
  // Kick off the DMA for the t=0 tile before entering the loop.
  if (wave == 0) {
    issue_x_tdm(x + (size_t)b0 * NIN,
                (unsigned)(unsigned long long)(size_t)(&xT[0][0]));
  }
#endif

  for (int t = 0; t < T_STEPS; ++t) {
    const int cur = t & 1;
#if !USE_TDM
    // Manual staging fallback: 16 rows x 256 cols, float4 per thread-chunk.
    {
      const float* xsrc = x + ((size_t)t * BATCH + (size_t)b0) * NIN;
      for (int i = tid; i < ROWS * (NIN / 4); i += BLOCK) {
        int row = i >> 6;              // NIN/4 == 64 chunks per row
        int c4  = (i & 63) << 2;
        const float4 v = *(const float4*)(xsrc + (size_t)row * NIN + c4);
        float* dst = &xT[cur][row * XSTRIDE + c4];
        dst[0] = v.x; dst[1] = v.y; dst[2] = v.z; dst[3] = v.w;
      }
    }
#else
    if (wave == 0) {
      __builtin_amdgcn_s_wait_tensorcnt(0);   // xT[cur] DMA complete
    }
#endif
    __syncthreads();   // xT[cur] (and spkT from prev step) visible; prev reads done

#if USE_TDM
    // Prefetch next step's x tile into the other buffer; overlaps all GEMMs.
    if (wave == 0 && t + 1 < T_STEPS) {
      issue_x_tdm(x + ((size_t)(t + 1) * BATCH + (size_t)b0) * NIN,
                  (unsigned)(unsigned long long)(size_t)(&xT[cur ^ 1][0]));
    }
#endif

    // ---- Phase 2: syn1 = ALPHA*syn1 + x_t@W1^T + spk1@Wrec^T --------------
    // Transposed GEMM: M = hidden n, N = batch b. A = weight rows (contiguous
    // global b64 loads), B = x / spike tiles from padded LDS.
    {
      const int nbase = wave * 64;
      #pragma unroll
      for (int nt = 0; nt < 4; ++nt) {
        const int nrow = nbase + nt * 16 + lm;
        const float* w1p = W1   + (size_t)nrow * NIN + 2 * lh;
        const float* wrp = Wrec + (size_t)nrow * NH  + 2 * lh;
        __builtin_prefetch(wrp, 0, 3);    // warm Wrec row while W1 GEMM runs
        v8f acc = syn[nt] * ALPHA_C;      // start accumulator at alpha*syn
        const float* xrow = &xT[cur][lm * XSTRIDE + 2 * lh];
        #pragma unroll 8
        for (int k0 = 0; k0 < NIN; k0 += 4) {
          v2f a = *(const v2f*)(w1p + k0);            // A[M=nrow, K=k0+2lh+{0,1}]
          v2f b; b.x = xrow[k0]; b.y = xrow[k0 + 1];  // B[K, N=b]
          acc = wmma4(a, b, acc);
        }
        #pragma unroll 8
        for (int k0 = 0; k0 < NH; k0 += 4) {
          v2f a = *(const v2f*)(wrp + k0);
          v2f b;
          b.x = spkT[(k0 + 2 * lh)     * SSTRIDE + lm];
          b.y = spkT[(k0 + 2 * lh + 1) * SSTRIDE + lm];
          acc = wmma4(a, b, acc);
        }
        syn[nt] = acc;
      }
    }
    __syncthreads();   // all waves done reading spkT(t-1)

    // ---- Phase 3: LIF1 (subtract reset, reset_delay=False) + publish spikes
    {
      const int nbase = wave * 64;
      #pragma unroll
      for (int nt = 0; nt < 4; ++nt) {
        v8f m = mem1[nt];
        const v8f s = syn[nt];
        #pragma unroll
        for (int r = 0; r < 8; ++r) {
          float mv    = m[r];
          float reset = (mv > THR_C) ? THR_C : 0.0f;  // fires on PREVIOUS mem
          mv = BETA_C * mv + s[r] - reset;
          float spk = (mv > THR_C) ? 1.0f : 0.0f;
          mv -= spk * THR_C;                          // immediate reset
          m[r] = mv;
          const int n = nbase + nt * 16 + r + 8 * lh; // C-layout -> hidden idx
          spkT[n * SSTRIDE + lm] = spk;
        }
        mem1[nt] = m;
      }
    }
    __syncthreads();   // spkT(t) visible to all waves

    // ---- Phase 4: mem2 = BETA*mem2 + spk1@W2^T - reset2; emit outputs -----
    // Un-flipped GEMM: M = batch b, N = out o. A = spike tile (LDS),
    // B = W2T rows (contiguous global), stores coalesced per C row.
    {
      const int obase = wave * 32;
      const size_t outb = (size_t)t * BATCH * NOUT;
      #pragma unroll
      for (int ot = 0; ot < 2; ++ot) {
        const int ocol = obase + ot * 16 + lm;
        v8f m = mem2[ot];
        v8f acc;
        #pragma unroll
        for (int r = 0; r < 8; ++r) {
          float mv = m[r];
          acc[r] = BETA_C * mv - ((mv > THR_C) ? THR_C : 0.0f);
        }
        const float* w2p = W2T + (size_t)(2 * lh) * NOUT + ocol;
        #pragma unroll 8
        for (int k0 = 0; k0 < NH; k0 += 4) {
          const int kb = k0 + 2 * lh;
          v2f a, b;
          a.x = spkT[kb       * SSTRIDE + lm];   // A[M=b=lm, K=kb+{0,1}]
          a.y = spkT[(kb + 1) * SSTRIDE + lm];
          b.x = w2p[(size_t)k0 * NOUT];          // B[K, N=ocol]
          b.y = w2p[(size_t)(k0 + 1) * NOUT];
          acc = wmma4(a, b, acc);
        }
        mem2[ot] = acc;
        #pragma unroll
        for (int r = 0; r < 8; ++r) {
          const int brow = b0 + r + 8 * lh;      // C-layout -> batch row
          const size_t idx = outb + (size_t)brow * NOUT + ocol;
          const float mv = acc[r];
          spk_out[idx] = (mv > THR_C) ? 1.0f : 0.0f;
          mem_out[idx] = mv;                     // recorded before next-step reset
        }
      }
    }
    // No end-of-loop barrier needed: the next iteration's top barrier orders
    // phase-4 spkT reads against the next LIF1 spike writes, and the x tiles
    // are double-buffered.
  }
}

extern "C" void kernel_launch(void* const* d_in, const int* in_sizes, int n_in,
                              void* d_out, int out_size, void* d_ws, size_t ws_size,
                              hipStream_t stream) {
  (void)in_sizes; (void)n_in; (void)out_size; (void)ws_size;
  const float* x    = (const float*)d_in[0];   // [T, B, NIN]
  const float* W1   = (const float*)d_in[1];   // [NH, NIN]
  const float* Wrec = (const float*)d_in[2];   // [NH, NH]
  const float* W2   = (const float*)d_in[3];   // [NOUT, NH]

  float* W2T     = (float*)d_ws;               // [NH, NOUT] = 512 KB scratch
  float* spk_out = (float*)d_out;              // [T, B, NOUT]
  float* mem_out = spk_out + (size_t)T_STEPS * BATCH * NOUT;

  transpose_w2_kernel<<<(NOUT * NH + 255) / 256, 256, 0, stream>>>(W2, W2T);
  snn_persistent_kernel<<<BATCH / ROWS, BLOCK, 0, stream>>>(
      x, W1, Wrec, W2T, spk_out, mem_out);
}